// GAT_73641509257451
// MI455X (gfx1250) — compile-verified
//
#include <hip/hip_runtime.h>
#include <cstdint>
#include <cstddef>

// ---------------------------------------------------------------------------
// 2-layer GAT (PyG GATConv semantics) for MI455X / gfx1250.
//  - GEMMs use native f32 WMMA: v_wmma_f32_16x16x4_f32 (exact f32 path).
//  - Edge softmax: 3-pass (atomicMax / atomicAdd denom / atomic scatter),
//    self-loops folded into per-node init kernels.
// ---------------------------------------------------------------------------

typedef float v2f __attribute__((ext_vector_type(2)));
typedef float v8f __attribute__((ext_vector_type(8)));

static __device__ __forceinline__ float lrelu(float x) { return x > 0.f ? x : 0.2f * x; }

// Monotone float<->uint mapping so float max == unsigned int max.
static __device__ __forceinline__ unsigned fenc(float f) {
    unsigned b = __float_as_uint(f);
    return (b & 0x80000000u) ? ~b : (b | 0x80000000u);
}
static __device__ __forceinline__ float fdec(unsigned k) {
    unsigned b = (k & 0x80000000u) ? (k & 0x7fffffffu) : ~k;
    return __uint_as_float(b);
}

// ---------------------------------------------------------------------------
// WMMA GEMM: C[M x NC] = A[M x K] * B[K x NC], row-major, f32.
// One wave -> 16 rows x NC cols. B staged in LDS in BK-deep chunks.
// A frag (16x4 f32): lane l holds M = l%16, K = (l/16)*2 + {vgpr0,vgpr1}.
// B frag (4x16 f32): lane l holds N = l%16, K = (l/16)*2 + {vgpr0,vgpr1}.
// C/D (16x16 f32): vgpr r, lanes0-15 -> M=r, lanes16-31 -> M=r+8, N=l%16.
// ---------------------------------------------------------------------------
template <int K, int NC, int BK>
__global__ void __launch_bounds__(256)
gat_wmma_gemm(const float* __restrict__ A, const float* __restrict__ B,
              float* __restrict__ C, int M) {
    __shared__ float sB[BK * NC];
    constexpr int NT = NC / 16;

    const int tid  = threadIdx.x;
    const int wave = tid >> 5;
    const int lane = tid & 31;
    const int wglobal = blockIdx.x * (blockDim.x >> 5) + wave;
    const int m0 = wglobal * 16;
    const bool store_ok = (m0 + 16 <= M);
    const int m0c = store_ok ? m0 : (M > 16 ? M - 16 : 0);  // clamp: keep EXEC all-1s for WMMA

    const int mrow = lane & 15;          // M (for A/C) or N (for B) within tile
    const int koff = (lane >> 4) << 1;   // 0 or 2: which K pair this half-wave owns
    const int rsel = (lane >> 4) << 3;   // 0 or 8: C row offset for this half-wave

    v8f acc[NT];
#pragma unroll
    for (int t = 0; t < NT; ++t) acc[t] = (v8f){0.f, 0.f, 0.f, 0.f, 0.f, 0.f, 0.f, 0.f};

    const float* arow = A + (size_t)(m0c + mrow) * K;

    for (int kc = 0; kc < K; kc += BK) {
        // Cooperative stage of B chunk into LDS (16B vectors, fully aligned).
        {
            const float4* gsrc = (const float4*)(B + (size_t)kc * NC);
            float4* sdst = (float4*)sB;
            for (int i = tid; i < (BK * NC) / 4; i += blockDim.x) sdst[i] = gsrc[i];
        }
        __syncthreads();

#pragma unroll 4
        for (int kk = 0; kk < BK; kk += 4) {
            const float2 av = *(const float2*)(arow + kc + kk + koff);
            v2f afrag; afrag.x = av.x; afrag.y = av.y;
#pragma unroll
            for (int t = 0; t < NT; ++t) {
                v2f bfrag;
                bfrag.x = sB[(kk + koff + 0) * NC + t * 16 + mrow];
                bfrag.y = sB[(kk + koff + 1) * NC + t * 16 + mrow];
                acc[t] = __builtin_amdgcn_wmma_f32_16x16x4_f32(
                    false, afrag, false, bfrag, (short)0, acc[t], false, false);
            }
        }
        __syncthreads();
    }

    if (store_ok) {
#pragma unroll
        for (int t = 0; t < NT; ++t)
#pragma unroll
            for (int r = 0; r < 8; ++r)
                C[(size_t)(m0 + r + rsel) * NC + t * 16 + mrow] = acc[t][r];
    }
}

// ---------------------------------------------------------------------------
// Layer-1 attention dots: a_src/a_dst [N,8] from h1 [N,64] (8 heads x 8 ch).
// ---------------------------------------------------------------------------
__global__ void gat_att8(const float* __restrict__ h1, const float* __restrict__ was,
                         const float* __restrict__ wad, float* __restrict__ asrc,
                         float* __restrict__ adst, int NH) {
    int idx = blockIdx.x * blockDim.x + threadIdx.x;
    if (idx >= NH) return;
    int n = idx >> 3, h = idx & 7;
    const float* r = h1 + (size_t)n * 64 + h * 8;
    float s = 0.f, d = 0.f;
#pragma unroll
    for (int c = 0; c < 8; ++c) { s += r[c] * was[h * 8 + c]; d += r[c] * wad[h * 8 + c]; }
    asrc[idx] = s; adst[idx] = d;
}

// Layer-2 attention dots: one wave per node, 128 channels, shuffle reduce.
__global__ void gat_att128(const float* __restrict__ h2, const float* __restrict__ was,
                           const float* __restrict__ wad, float* __restrict__ asrc,
                           float* __restrict__ adst, int N) {
    int wave = threadIdx.x >> 5, lane = threadIdx.x & 31;
    int n = blockIdx.x * (blockDim.x >> 5) + wave;
    if (n >= N) return;
    const float4 hv = *(const float4*)(h2 + (size_t)n * 128 + lane * 4);
    const float4 sv = *(const float4*)(was + lane * 4);
    const float4 dv = *(const float4*)(wad + lane * 4);
    float s = hv.x * sv.x + hv.y * sv.y + hv.z * sv.z + hv.w * sv.w;
    float d = hv.x * dv.x + hv.y * dv.y + hv.z * dv.z + hv.w * dv.w;
#pragma unroll
    for (int m = 16; m >= 1; m >>= 1) { s += __shfl_xor(s, m, 32); d += __shfl_xor(d, m, 32); }
    if (lane == 0) { asrc[n] = s; adst[n] = d; }
}

// Init amax with the self-loop alpha (covers isolated nodes too).
__global__ void gat_max_init(const float* __restrict__ asrc, const float* __restrict__ adst,
                             unsigned* __restrict__ amax, int cnt) {
    int i = blockIdx.x * blockDim.x + threadIdx.x;
    if (i >= cnt) return;
    amax[i] = fenc(lrelu(asrc[i] + adst[i]));
}

template <int H>
__global__ void gat_edge_max(const long long* __restrict__ esrc, const long long* __restrict__ edst,
                             const float* __restrict__ asrc, const float* __restrict__ adst,
                             unsigned* __restrict__ amax, int E) {
    int e = blockIdx.x * blockDim.x + threadIdx.x;
    if (e >= E) return;
    int s = (int)esrc[e], d = (int)edst[e];
#pragma unroll
    for (int h = 0; h < H; ++h)
        atomicMax(&amax[d * H + h], fenc(lrelu(asrc[s * H + h] + adst[d * H + h])));
}

// denom init = self-loop contribution exp(self_alpha - amax).
__global__ void gat_den_init(const float* __restrict__ asrc, const float* __restrict__ adst,
                             const unsigned* __restrict__ amax, float* __restrict__ den, int cnt) {
    int i = blockIdx.x * blockDim.x + threadIdx.x;
    if (i >= cnt) return;
    den[i] = expf(lrelu(asrc[i] + adst[i]) - fdec(amax[i]));
}

template <int H>
__global__ void gat_edge_sum(const long long* __restrict__ esrc, const long long* __restrict__ edst,
                             const float* __restrict__ asrc, const float* __restrict__ adst,
                             const unsigned* __restrict__ amax, float* __restrict__ den, int E) {
    int e = blockIdx.x * blockDim.x + threadIdx.x;
    if (e >= E) return;
    int s = (int)esrc[e], d = (int)edst[e];
#pragma unroll
    for (int h = 0; h < H; ++h)
        atomicAdd(&den[d * H + h],
                  expf(lrelu(asrc[s * H + h] + adst[d * H + h]) - fdec(amax[d * H + h])));
}

// out init = bias + self-loop message (fully overwrites the accumulator).
template <int H, int C>
__global__ void gat_out_init(const float* __restrict__ hfeat, const float* __restrict__ bias,
                             const float* __restrict__ asrc, const float* __restrict__ adst,
                             const unsigned* __restrict__ amax, const float* __restrict__ den,
                             float* __restrict__ outacc, int total) {
    int idx = blockIdx.x * blockDim.x + threadIdx.x;
    if (idx >= total) return;
    int c = idx % (H * C);
    int n = idx / (H * C);
    int h = c / C;
    float sa = lrelu(asrc[n * H + h] + adst[n * H + h]);
    float w = expf(sa - fdec(amax[n * H + h])) / (den[n * H + h] + 1e-16f);
    outacc[idx] = bias[c] + hfeat[idx] * w;
}

template <int H, int C>
__global__ void gat_edge_scatter(const long long* __restrict__ esrc, const long long* __restrict__ edst,
                                 const float* __restrict__ hfeat, const float* __restrict__ asrc,
                                 const float* __restrict__ adst, const unsigned* __restrict__ amax,
                                 const float* __restrict__ den, float* __restrict__ outacc, int total) {
    int idx = blockIdx.x * blockDim.x + threadIdx.x;
    if (idx >= total) return;
    int c = idx % (H * C);
    int e = idx / (H * C);
    int h = c / C;
    int s = (int)esrc[e], d = (int)edst[e];
    float a = lrelu(asrc[s * H + h] + adst[d * H + h]);
    float w = expf(a - fdec(amax[d * H + h])) / (den[d * H + h] + 1e-16f);
    atomicAdd(&outacc[(size_t)d * (H * C) + c], hfeat[(size_t)s * (H * C) + c] * w);
}

__global__ void gat_elu(float* __restrict__ v, int cnt) {
    int i = blockIdx.x * blockDim.x + threadIdx.x;
    if (i >= cnt) return;
    float x = v[i];
    v[i] = x > 0.f ? x : expm1f(x);
}

// ---------------------------------------------------------------------------

static inline int cdiv(long long a, long long b) { return (int)((a + b - 1) / b); }

extern "C" void kernel_launch(void* const* d_in, const int* in_sizes, int n_in,
                              void* d_out, int out_size, void* d_ws, size_t ws_size,
                              hipStream_t stream) {
    const float*     x    = (const float*)d_in[0];
    const long long* ei   = (const long long*)d_in[1];  // int64 [2,E]
    const float*     W1   = (const float*)d_in[2];      // [512,64]
    const float*     as1w = (const float*)d_in[3];      // [8,8]
    const float*     ad1w = (const float*)d_in[4];
    const float*     b1   = (const float*)d_in[5];      // [64]
    const float*     W2   = (const float*)d_in[6];      // [64,128]
    const float*     as2w = (const float*)d_in[7];      // [1,128]
    const float*     ad2w = (const float*)d_in[8];
    const float*     b2   = (const float*)d_in[9];      // [128]
    float* out = (float*)d_out;                         // [N,128]

    const int N = in_sizes[0] / 512;
    const int E = in_sizes[1] / 2;
    const long long* esrc = ei;
    const long long* edst = ei + E;

    // Workspace layout (floats). Every region is fully written before read
    // on every call, so no init / no cross-call state.
    float* ws = (float*)d_ws;
    size_t o = 0;
    float*    h1     = ws + o; o += (size_t)N * 64;
    float*    pasrc1 = ws + o; o += (size_t)N * 8;
    float*    padst1 = ws + o; o += (size_t)N * 8;
    unsigned* amax1  = (unsigned*)(ws + o); o += (size_t)N * 8;
    float*    den1   = ws + o; o += (size_t)N * 8;
    float*    acc1   = ws + o; o += (size_t)N * 64;   // layer-1 out, then h_mid (in-place ELU)
    float*    h2     = ws + o; o += (size_t)N * 128;
    float*    pasrc2 = ws + o; o += (size_t)N;
    float*    padst2 = ws + o; o += (size_t)N;
    unsigned* amax2  = (unsigned*)(ws + o); o += (size_t)N;
    float*    den2   = ws + o; o += (size_t)N;

    const int TB = 256;
    const int rowTiles = cdiv(N, 16);
    const int gemmBlocks = cdiv(rowTiles, 8);  // 8 waves/block

    // ---- Layer 1: h1 = x @ W1  (WMMA f32) ----
    gat_wmma_gemm<512, 64, 64><<<gemmBlocks, TB, 0, stream>>>(x, W1, h1, N);
    gat_att8<<<cdiv((long long)N * 8, TB), TB, 0, stream>>>(h1, as1w, ad1w, pasrc1, padst1, N * 8);
    gat_max_init<<<cdiv((long long)N * 8, TB), TB, 0, stream>>>(pasrc1, padst1, amax1, N * 8);
    gat_edge_max<8><<<cdiv(E, TB), TB, 0, stream>>>(esrc, edst, pasrc1, padst1, amax1, E);
    gat_den_init<<<cdiv((long long)N * 8, TB), TB, 0, stream>>>(pasrc1, padst1, amax1, den1, N * 8);
    gat_edge_sum<8><<<cdiv(E, TB), TB, 0, stream>>>(esrc, edst, pasrc1, padst1, amax1, den1, E);
    gat_out_init<8, 8><<<cdiv((long long)N * 64, TB), TB, 0, stream>>>(
        h1, b1, pasrc1, padst1, amax1, den1, acc1, N * 64);
    gat_edge_scatter<8, 8><<<cdiv((long long)E * 64, TB), TB, 0, stream>>>(
        esrc, edst, h1, pasrc1, padst1, amax1, den1, acc1, E * 64);
    gat_elu<<<cdiv((long long)N * 64, TB), TB, 0, stream>>>(acc1, N * 64);

    // ---- Layer 2: h2 = elu_out @ W2  (WMMA f32) ----
    gat_wmma_gemm<64, 128, 64><<<gemmBlocks, TB, 0, stream>>>(acc1, W2, h2, N);
    gat_att128<<<cdiv(N, 8), TB, 0, stream>>>(h2, as2w, ad2w, pasrc2, padst2, N);
    gat_max_init<<<cdiv(N, TB), TB, 0, stream>>>(pasrc2, padst2, amax2, N);
    gat_edge_max<1><<<cdiv(E, TB), TB, 0, stream>>>(esrc, edst, pasrc2, padst2, amax2, E);
    gat_den_init<<<cdiv(N, TB), TB, 0, stream>>>(pasrc2, padst2, amax2, den2, N);
    gat_edge_sum<1><<<cdiv(E, TB), TB, 0, stream>>>(esrc, edst, pasrc2, padst2, amax2, den2, E);
    gat_out_init<1, 128><<<cdiv((long long)N * 128, TB), TB, 0, stream>>>(
        h2, b2, pasrc2, padst2, amax2, den2, out, N * 128);
    gat_edge_scatter<1, 128><<<cdiv((long long)E * 128, TB), TB, 0, stream>>>(
        esrc, edst, h2, pasrc2, padst2, amax2, den2, out, E * 128);
}